// UpdateSuperpointsModule_7146825581107
// MI455X (gfx1250) — compile-verified
//
#include <hip/hip_runtime.h>
#include <hip/hip_bf16.h>

// ---------------- problem constants (match reference setup_inputs) ----------
#define S0   512      // num_segments0
#define P0   1024     // points_per_segment0
#define S1   16384    // num_segments1
#define SPL0 32       // segments_per_level0
#define NB   96       // 3 neighbor segments * SPL0
#define C    64       // feature channels
#define KNN  5        // module K
#define NPT  (S0 * P0)

// output layout (flat, reference return order, all f32)
#define OFF_ASSIGN   0
#define OFF_SPFEAT   (OFF_ASSIGN + NPT)              //   524288
#define OFF_SPCOORD  (OFF_SPFEAT + S1 * C)           //  1572864
#define OFF_PTSFEAT  (OFF_SPCOORD + S1 * 3)          //  1622016
#define OFF_HILFEAT  (OFF_PTSFEAT + NPT * C)         // 35176448

typedef __attribute__((ext_vector_type(2))) float v2f;
typedef __attribute__((ext_vector_type(8))) float v8f;

// ---------------------------------------------------------------------------
// Fused assignment kernel: one block (8 waves) per level-0 segment.
// LDS-stages the 96 candidate superpoints (coords+|n|^2, 64-ch features,
// global ids), then each wave computes point-x-neighbor dot blocks with
// v_wmma_f32_16x16x4_f32 (coords zero-padded K=3->4), derives squared
// distances, does stable top-5 per point, 5x64 feature dots + argmax,
// writes assignment and atomically accumulates scatter-mean partials.
//
// The 16x16 WMMA result is transposed through a small per-wave LDS tile.
// DS ops from one wave are executed in order by hardware; we add an explicit
// s_wait_dscnt 0 + compiler memory barrier between the spill and the reload
// (no volatile -> accesses stay in the LDS address space as ds_store/ds_load).
// ---------------------------------------------------------------------------
__global__ __launch_bounds__(256)
void assign_kernel(const float* __restrict__ sp_feat,    // [S1,C]
                   const float* __restrict__ sp_coord,   // [S1,3]
                   const float* __restrict__ pts_feat,   // [S0,P0,C]
                   const float* __restrict__ pts_coord,  // [S0,P0,3]
                   const float* __restrict__ hil_coord,  // [N,3]
                   const int*   __restrict__ l0l1,       // [S0,SPL0]
                   float* __restrict__ out_assign,       // [N] (float-encoded int)
                   float* __restrict__ ws_sum,           // [S1*3]
                   float* __restrict__ ws_cnt)           // [S1]
{
    const int s    = blockIdx.x;
    const int tid  = threadIdx.x;
    const int lane = tid & 31;
    const int w    = tid >> 5;

    __shared__ float  nbf[NB * C];     // neighbor superpoint features
    __shared__ float4 nbc[NB];         // x, y, z, |n|^2
    __shared__ int    nsp[NB];         // global superpoint index
    __shared__ float  d2t[8][16 * 20]; // per-wave dot tile, addr = col*20 + M

    // --- stage neighbor superpoint data -----------------------------------
    for (int j = tid; j < NB; j += 256) {
        int ns = s + (j / SPL0) - 1;
        ns = ns < 0 ? 0 : (ns > S0 - 1 ? S0 - 1 : ns);
        int sp = l0l1[ns * SPL0 + (j % SPL0)];
        nsp[j] = sp;
        float x = sp_coord[sp * 3 + 0];
        float y = sp_coord[sp * 3 + 1];
        float z = sp_coord[sp * 3 + 2];
        nbc[j] = make_float4(x, y, z, x * x + y * y + z * z);
    }
    for (int idx = tid; idx < NB * C; idx += 256) {
        int j  = idx >> 6;                      // neighbor row
        int ns = s + (j / SPL0) - 1;
        ns = ns < 0 ? 0 : (ns > S0 - 1 ? S0 - 1 : ns);
        int sp = l0l1[ns * SPL0 + (j % SPL0)];  // recompute: no extra barrier
        nbf[idx] = sp_feat[(size_t)sp * C + (idx & (C - 1))];
    }
    __syncthreads();

    const int row = lane & 15;   // point row (reader) / neighbor col (writer)
    const int hi  = lane >> 4;   // half-wave select (K split for f32 WMMA)
    float* dbuf = d2t[w];

    // 1024 points / 8 waves = 8 tiles of 16 points per wave
    for (int tt = 0; tt < 8; ++tt) {
        const int pb = (w * 8 + tt) * 16;
        const int p  = pb + row;
        const int gp = s * P0 + p;

        float px = pts_coord[(size_t)gp * 3 + 0];
        float py = pts_coord[(size_t)gp * 3 + 1];
        float pz = pts_coord[(size_t)gp * 3 + 2];
        float pn2 = px * px + py * py + pz * pz;

        // A (16x4 f32): lanes 0-15 hold K0/K1, lanes 16-31 hold K2/K3(pad 0)
        v2f a;
        a.x = hi ? pz   : px;
        a.y = hi ? 0.0f : py;

        float bd[KNN]; int bi[KNN];
#pragma unroll
        for (int k = 0; k < KNN; ++k) { bd[k] = 3.4e38f; bi[k] = 0x7fffffff; }

        for (int n = 0; n < NB / 16; ++n) {
            float4 nc = nbc[n * 16 + row];
            // B (4x16 f32): same K split across half-waves as A
            v2f b;
            b.x = hi ? nc.z : nc.x;
            b.y = hi ? 0.0f : nc.y;

            v8f cacc = {};
            cacc = __builtin_amdgcn_wmma_f32_16x16x4_f32(
                false, a, false, b, (short)0, cacc, false, false);

            // C layout: VGPR r holds M=r (lanes 0-15) / M=r+8 (lanes 16-31),
            // N = lane&15. Column-major spill: lane's 8 values contiguous ->
            // two ds_store_b128. addr(col, M) = col*20 + M.
            float4* dst = (float4*)(dbuf + row * 20 + hi * 8);
            dst[0] = make_float4(cacc[0], cacc[1], cacc[2], cacc[3]);
            dst[1] = make_float4(cacc[4], cacc[5], cacc[6], cacc[7]);

            // intra-wave DS ordering: hardware keeps DS in order; wait covers
            // VGPR return, memory clobber stops compiler reordering.
            asm volatile("s_wait_dscnt 0x0" ::: "memory");

            // stable top-5 scan of the 16 fresh neighbor distances
            if (lane < 16) {
#pragma unroll
                for (int jn = 0; jn < 16; ++jn) {
                    int   nj  = n * 16 + jn;
                    float dot = dbuf[jn * 20 + row];
                    float d2  = (pn2 - 2.0f * dot) + nbc[nj].w;
                    if (d2 < bd[KNN - 1]) {
                        int pos = KNN - 1;
#pragma unroll
                        for (int k = KNN - 2; k >= 0; --k) {
                            if (d2 < bd[k]) {
                                bd[k + 1] = bd[k];
                                bi[k + 1] = bi[k];
                                pos = k;
                            }
                        }
                        bd[pos] = d2; bi[pos] = nj;
                    }
                }
            }
        }

        if (lane < 16) {
            // 5 feature dot products (softmax is monotone -> argmax of dots)
            const float* pf = pts_feat + (size_t)gp * C;
            float dk[KNN];
#pragma unroll
            for (int k = 0; k < KNN; ++k) dk[k] = 0.0f;
            for (int c4 = 0; c4 < C; c4 += 4) {
                float4 f = *(const float4*)(pf + c4);
#pragma unroll
                for (int k = 0; k < KNN; ++k) {
                    const float* nr = &nbf[bi[k] * C + c4];
                    dk[k] += f.x * nr[0] + f.y * nr[1] + f.z * nr[2] + f.w * nr[3];
                }
            }
            int bestk = 0; float bv = dk[0];
#pragma unroll
            for (int k = 1; k < KNN; ++k)
                if (dk[k] > bv) { bv = dk[k]; bestk = k; }  // first-max tie break

            int assigned = nsp[bi[bestk]];
            out_assign[gp] = (float)assigned;

            float hx = hil_coord[(size_t)gp * 3 + 0];
            float hy = hil_coord[(size_t)gp * 3 + 1];
            float hz = hil_coord[(size_t)gp * 3 + 2];
            atomicAdd(&ws_sum[assigned * 3 + 0], hx);
            atomicAdd(&ws_sum[assigned * 3 + 1], hy);
            atomicAdd(&ws_sum[assigned * 3 + 2], hz);
            atomicAdd(&ws_cnt[assigned], 1.0f);
        }
    }
}

// ---------------- layernorm: one wave per 64-ch row ------------------------
__global__ __launch_bounds__(256)
void ln_kernel(const float* __restrict__ f, const float* __restrict__ wgt,
               const float* __restrict__ bias, float* __restrict__ out)
{
    int w    = threadIdx.x >> 5;
    int lane = threadIdx.x & 31;
    int r    = blockIdx.x * 8 + w;      // 2048 blocks cover 16384 rows
    const float* x = f + (size_t)r * C;
    float x0 = x[lane], x1 = x[lane + 32];
    float sum = x0 + x1;
#pragma unroll
    for (int m = 16; m > 0; m >>= 1) sum += __shfl_xor(sum, m, 32);
    float mu = sum * (1.0f / 64.0f);
    float c0 = x0 - mu, c1 = x1 - mu;
    float v = c0 * c0 + c1 * c1;
#pragma unroll
    for (int m = 16; m > 0; m >>= 1) v += __shfl_xor(v, m, 32);
    float rs = rsqrtf(v * (1.0f / 64.0f) + 1e-5f);
    float* o = out + (size_t)r * C;
    o[lane]      = c0 * rs * wgt[lane]      + bias[lane];
    o[lane + 32] = c1 * rs * wgt[lane + 32] + bias[lane + 32];
}

// ---------------- workspace zero (deterministic each call) -----------------
__global__ void zero_kernel(float* __restrict__ p, int n)
{
    int i = blockIdx.x * 256 + threadIdx.x;
    if (i < n) p[i] = 0.0f;
}

// ---------------- scatter-mean finalize ------------------------------------
__global__ void mean_kernel(const float* __restrict__ sum,
                            const float* __restrict__ cnt,
                            float* __restrict__ out)
{
    int i = blockIdx.x * 256 + threadIdx.x;
    if (i < S1) {
        float c = fmaxf(cnt[i], 1.0f);
        out[i * 3 + 0] = sum[i * 3 + 0] / c;
        out[i * 3 + 1] = sum[i * 3 + 1] / c;
        out[i * 3 + 2] = sum[i * 3 + 2] / c;
    }
}

// ---------------- read-once / write-twice copy of rawPoint_feat ------------
__global__ __launch_bounds__(256)
void copy2_kernel(const float4* __restrict__ src,
                  float4* __restrict__ d1,
                  float4* __restrict__ d2, int n4)
{
    int i      = blockIdx.x * blockDim.x + threadIdx.x;
    int stride = gridDim.x * blockDim.x;
    for (; i < n4; i += stride) {
        float4 v = src[i];
        d1[i] = v;
        d2[i] = v;
    }
}

extern "C" void kernel_launch(void* const* d_in, const int* in_sizes, int n_in,
                              void* d_out, int out_size, void* d_ws, size_t ws_size,
                              hipStream_t stream)
{
    const float* sp_feat   = (const float*)d_in[0];
    const float* sp_coord  = (const float*)d_in[1];
    const float* raw_feat  = (const float*)d_in[2];
    const float* hil_coord = (const float*)d_in[3];
    const float* pts_feat  = (const float*)d_in[4];
    const float* pts_coord = (const float*)d_in[5];
    const float* norm_w    = (const float*)d_in[6];
    const float* norm_b    = (const float*)d_in[7];
    const int*   l0l1      = (const int*)d_in[8];

    float* out         = (float*)d_out;
    float* out_assign  = out + OFF_ASSIGN;
    float* out_spfeat  = out + OFF_SPFEAT;
    float* out_spcoord = out + OFF_SPCOORD;
    float* out_ptsfeat = out + OFF_PTSFEAT;
    float* out_hilfeat = out + OFF_HILFEAT;

    float* ws_sum = (float*)d_ws;          // S1*3 floats
    float* ws_cnt = ws_sum + S1 * 3;       // S1 floats

    zero_kernel<<<(S1 * 4 + 255) / 256, 256, 0, stream>>>(ws_sum, S1 * 4);

    assign_kernel<<<S0, 256, 0, stream>>>(sp_feat, sp_coord, pts_feat, pts_coord,
                                          hil_coord, l0l1, out_assign,
                                          ws_sum, ws_cnt);

    ln_kernel<<<S1 / 8, 256, 0, stream>>>(sp_feat, norm_w, norm_b, out_spfeat);

    mean_kernel<<<(S1 + 255) / 256, 256, 0, stream>>>(ws_sum, ws_cnt, out_spcoord);

    copy2_kernel<<<4096, 256, 0, stream>>>((const float4*)raw_feat,
                                           (float4*)out_ptsfeat,
                                           (float4*)out_hilfeat,
                                           (NPT * C) / 4);
}